// Node_Classification2_32220844654962
// MI455X (gfx1250) — compile-verified
//
#include <hip/hip_runtime.h>
#include <hip/hip_bf16.h>
#include <math.h>

// ---------------------------------------------------------------------------
// GATv2 stack, MI455X (gfx1250, wave32). fp32 end-to-end; GEMMs via
// V_WMMA_F32_16X16X4_F32 (exact fp32 matrix path). Edge phase is L2-resident
// (all node arrays ~12.8MB << 192MB L2) with float4 gathers + f32 atomics.
// ---------------------------------------------------------------------------

typedef __attribute__((ext_vector_type(2))) float v2f;
typedef __attribute__((ext_vector_type(8))) float v8f;

#define HID 64
#define NEG_SLOPE 0.2f
#define LN_EPS 1e-5f

// ---- order-preserving float <-> uint key (for atomic segment-max) ----------
__device__ __forceinline__ unsigned fkey(float f) {
    unsigned b = __float_as_uint(f);
    return (b & 0x80000000u) ? ~b : (b | 0x80000000u);
}
__device__ __forceinline__ float funkey(unsigned k) {
    unsigned b = (k & 0x80000000u) ? (k ^ 0x80000000u) : ~k;
    return __uint_as_float(b);
}
#define KEY_NEG_INF 0x007FFFFFu   // fkey(-inf)

// ---------------------------------------------------------------------------
// GEMM: Y[nrows, nout] = X[nrows, 64] @ W[64, nout] + bias
// 128 threads = 4 waves; each wave owns 16 rows, NTILES 16-col output tiles.
// fp32 WMMA: A is 16x4 (v2f/lane), B is 4x16 (v2f/lane), C/D 16x16 (v8f).
// ---------------------------------------------------------------------------
template <int NTILES>
__global__ void gemm64_wmma(const float* __restrict__ X,
                            const float* __restrict__ W,
                            const float* __restrict__ bias,
                            float* __restrict__ Y,
                            int nrows, int nout) {
    __shared__ float lds_w[64][64];

    int tid = threadIdx.x;
    // stage W (zero-padded to 64 cols) into LDS
    for (int idx = tid; idx < 64 * 64; idx += 128) {
        int r = idx >> 6, c = idx & 63;
        lds_w[r][c] = (c < nout) ? W[r * nout + c] : 0.0f;
    }
    __syncthreads();

    int lane  = tid & 31;
    int wave  = tid >> 5;
    int half  = lane >> 4;     // 0: lanes 0-15, 1: lanes 16-31
    int l15   = lane & 15;
    int rbase = blockIdx.x * 64 + wave * 16;

    int arow = rbase + l15;
    if (arow >= nrows) arow = nrows - 1;          // clamp (EXEC stays all-1)
    const float* xrow = X + (size_t)arow * 64;

    v8f acc[NTILES];
#pragma unroll
    for (int t = 0; t < NTILES; ++t)
#pragma unroll
        for (int j = 0; j < 8; ++j) acc[t][j] = 0.0f;

#pragma unroll
    for (int k0 = 0; k0 < 64; k0 += 4) {
        // A fragment: lane<16 -> K = k0,k0+1 ; lane>=16 -> K = k0+2,k0+3
        const float* ap = xrow + k0 + 2 * half;
        v2f a;
        a.x = ap[0];
        a.y = ap[1];
#pragma unroll
        for (int t = 0; t < NTILES; ++t) {
            v2f b;
            b.x = lds_w[k0 + 2 * half + 0][t * 16 + l15];
            b.y = lds_w[k0 + 2 * half + 1][t * 16 + l15];
            acc[t] = __builtin_amdgcn_wmma_f32_16x16x4_f32(
                false, a, false, b, (short)0, acc[t], false, false);
        }
    }

    // D layout: VGPR i -> M = 8*half + i, N = tile*16 + l15
#pragma unroll
    for (int t = 0; t < NTILES; ++t) {
        int col = t * 16 + l15;
        float bv = (col < nout) ? bias[col] : 0.0f;
#pragma unroll
        for (int i = 0; i < 8; ++i) {
            int row = rbase + 8 * half + i;
            if (row < nrows && col < nout)
                Y[(size_t)row * nout + col] = acc[t][i] + bv;
        }
    }
}

// ---------------------------------------------------------------------------
// per-layer init: t = 0, denom = 0, mx = key(-inf)
// ---------------------------------------------------------------------------
__global__ void init_layer(float* __restrict__ t, float* __restrict__ denom,
                           unsigned* __restrict__ mx, int n) {
    int i = blockIdx.x * blockDim.x + threadIdx.x;
    if (i < n * 16) {
        float4 z = make_float4(0.f, 0.f, 0.f, 0.f);
        ((float4*)t)[i] = z;
    }
    if (i < n) {
        denom[i] = 0.0f;
        mx[i]    = KEY_NEG_INF;
    }
}

// ---------------------------------------------------------------------------
// pass 1: score[e] = sum_h leakyrelu(fs[src] + fd[dst]) * attn ; atomicMax dst
// ---------------------------------------------------------------------------
__global__ void edge_score(const float* __restrict__ fs,
                           const float* __restrict__ fd,
                           const int* __restrict__ src,
                           const int* __restrict__ dst,
                           const float* __restrict__ attn,
                           float* __restrict__ score,
                           unsigned* __restrict__ mx, int E) {
    int e = blockIdx.x * blockDim.x + threadIdx.x;
    if (e >= E) return;
    int s = src[e], d = dst[e];
    const float4* ps = (const float4*)(fs + (size_t)s * HID);
    const float4* pd = (const float4*)(fd + (size_t)d * HID);
    const float4* pa = (const float4*)attn;
    float acc = 0.0f;
#pragma unroll
    for (int i = 0; i < HID / 4; ++i) {
        float4 a = ps[i], b = pd[i], w = pa[i];
        float x0 = a.x + b.x, x1 = a.y + b.y, x2 = a.z + b.z, x3 = a.w + b.w;
        x0 = x0 > 0.f ? x0 : x0 * NEG_SLOPE;
        x1 = x1 > 0.f ? x1 : x1 * NEG_SLOPE;
        x2 = x2 > 0.f ? x2 : x2 * NEG_SLOPE;
        x3 = x3 > 0.f ? x3 : x3 * NEG_SLOPE;
        acc += x0 * w.x + x1 * w.y + x2 * w.z + x3 * w.w;
    }
    score[e] = acc;
    atomicMax(&mx[d], fkey(acc));
}

// ---------------------------------------------------------------------------
// pass 2: ex = exp(score - mx[dst]); denom[dst] += ex; score <- ex
// ---------------------------------------------------------------------------
__global__ void edge_exp(float* __restrict__ score,
                         const unsigned* __restrict__ mx,
                         const int* __restrict__ dst,
                         float* __restrict__ denom, int E) {
    int e = blockIdx.x * blockDim.x + threadIdx.x;
    if (e >= E) return;
    int d = dst[e];
    float ex = expf(score[e] - funkey(mx[d]));
    score[e] = ex;
    atomicAdd(&denom[d], ex);
}

// ---------------------------------------------------------------------------
// pass 3: t[dst] += (ex/denom[dst]) * fs[src]
// ---------------------------------------------------------------------------
__global__ void edge_scatter(const float* __restrict__ score,
                             const float* __restrict__ denom,
                             const float* __restrict__ fs,
                             const int* __restrict__ src,
                             const int* __restrict__ dst,
                             float* __restrict__ t, int E) {
    int e = blockIdx.x * blockDim.x + threadIdx.x;
    if (e >= E) return;
    int s = src[e], d = dst[e];
    float alpha = score[e] / denom[d];
    const float4* ps = (const float4*)(fs + (size_t)s * HID);
    float* pt = t + (size_t)d * HID;
#pragma unroll
    for (int i = 0; i < HID / 4; ++i) {
        float4 v = ps[i];
        atomicAdd(pt + 4 * i + 0, v.x * alpha);
        atomicAdd(pt + 4 * i + 1, v.y * alpha);
        atomicAdd(pt + 4 * i + 2, v.z * alpha);
        atomicAdd(pt + 4 * i + 3, v.w * alpha);
    }
}

// ---------------------------------------------------------------------------
// node update: h += gelu(layernorm(t) * gamma + beta)   (exact gelu via erf)
// ---------------------------------------------------------------------------
__global__ void node_update(float* __restrict__ h, const float* __restrict__ t,
                            const float* __restrict__ gamma,
                            const float* __restrict__ beta, int n) {
    int i = blockIdx.x * blockDim.x + threadIdx.x;
    if (i >= n) return;
    const float4* pt = (const float4*)(t + (size_t)i * HID);
    float4 v[HID / 4];
    float sum = 0.0f;
#pragma unroll
    for (int j = 0; j < HID / 4; ++j) {
        v[j] = pt[j];
        sum += v[j].x + v[j].y + v[j].z + v[j].w;
    }
    float mean = sum * (1.0f / HID);
    float var = 0.0f;
#pragma unroll
    for (int j = 0; j < HID / 4; ++j) {
        float d0 = v[j].x - mean, d1 = v[j].y - mean;
        float d2 = v[j].z - mean, d3 = v[j].w - mean;
        var += d0 * d0 + d1 * d1 + d2 * d2 + d3 * d3;
    }
    var *= (1.0f / HID);
    float inv = rsqrtf(var + LN_EPS);

    float4* ph = (float4*)(h + (size_t)i * HID);
    const float4* pg = (const float4*)gamma;
    const float4* pb = (const float4*)beta;
#pragma unroll
    for (int j = 0; j < HID / 4; ++j) {
        float4 g = pg[j], b = pb[j], hv = ph[j];
        float x0 = (v[j].x - mean) * inv * g.x + b.x;
        float x1 = (v[j].y - mean) * inv * g.y + b.y;
        float x2 = (v[j].z - mean) * inv * g.z + b.z;
        float x3 = (v[j].w - mean) * inv * g.w + b.w;
        const float k = 0.70710678118654752f;
        hv.x += 0.5f * x0 * (1.0f + erff(x0 * k));
        hv.y += 0.5f * x1 * (1.0f + erff(x1 * k));
        hv.z += 0.5f * x2 * (1.0f + erff(x2 * k));
        hv.w += 0.5f * x3 * (1.0f + erff(x3 * k));
        ph[j] = hv;
    }
}

// ---------------------------------------------------------------------------
extern "C" void kernel_launch(void* const* d_in, const int* in_sizes, int n_in,
                              void* d_out, int out_size, void* d_ws, size_t ws_size,
                              hipStream_t stream) {
    const float* nodes = (const float*)d_in[0];
    const int*   src   = (const int*)d_in[1];
    const int*   dst   = (const int*)d_in[2];
    const float* w_in  = (const float*)d_in[3];
    const float* b_in  = (const float*)d_in[4];
    const float* w_src = (const float*)d_in[5];
    const float* b_src = (const float*)d_in[6];
    const float* w_dst = (const float*)d_in[7];
    const float* b_dst = (const float*)d_in[8];
    const float* attn  = (const float*)d_in[9];
    const float* gamma = (const float*)d_in[10];
    const float* beta  = (const float*)d_in[11];
    const float* w_out = (const float*)d_in[12];
    const float* b_out = (const float*)d_in[13];

    const int N = in_sizes[0] / HID;   // 50000
    const int E = in_sizes[1];         // 500000
    const int NL = 6;

    // workspace carve-up (fp32)
    float* ws   = (float*)d_ws;
    float* h    = ws;  ws += (size_t)N * HID;
    float* fs   = ws;  ws += (size_t)N * HID;
    float* fd   = ws;  ws += (size_t)N * HID;
    float* tbuf = ws;  ws += (size_t)N * HID;
    float* score = ws; ws += E;
    float* denom = ws; ws += N;
    unsigned* mx = (unsigned*)ws;

    const int ggrid  = (N + 63) / 64;
    const int eblk   = 256;
    const int egrid  = (E + eblk - 1) / eblk;
    const int ngrid  = (N + 255) / 256;
    const int igrid  = (N * 16 + 255) / 256;

    // input projection
    gemm64_wmma<4><<<ggrid, 128, 0, stream>>>(nodes, w_in, b_in, h, N, HID);

    for (int L = 0; L < NL; ++L) {
        gemm64_wmma<4><<<ggrid, 128, 0, stream>>>(h, w_src + (size_t)L * HID * HID,
                                                  b_src + (size_t)L * HID, fs, N, HID);
        gemm64_wmma<4><<<ggrid, 128, 0, stream>>>(h, w_dst + (size_t)L * HID * HID,
                                                  b_dst + (size_t)L * HID, fd, N, HID);
        init_layer<<<igrid, 256, 0, stream>>>(tbuf, denom, mx, N);
        edge_score<<<egrid, eblk, 0, stream>>>(fs, fd, src, dst,
                                               attn + (size_t)L * HID, score, mx, E);
        edge_exp<<<egrid, eblk, 0, stream>>>(score, mx, dst, denom, E);
        edge_scatter<<<egrid, eblk, 0, stream>>>(score, denom, fs, src, dst, tbuf, E);
        node_update<<<ngrid, 256, 0, stream>>>(h, tbuf, gamma + (size_t)L * HID,
                                               beta + (size_t)L * HID, N);
    }

    // output projection (nout = 8, padded to one 16-col WMMA tile)
    gemm64_wmma<1><<<ggrid, 128, 0, stream>>>(h, w_out, b_out, (float*)d_out, N, 8);
}